// SNATConv_49246095016139
// MI455X (gfx1250) — compile-verified
//
#include <hip/hip_runtime.h>
#include <hip/hip_bf16.h>

// ---------------- problem constants ----------------
#define N_NODES 100000
#define E_EDGES 1600000
#define DIN     256
#define HT      512
#define NHEAD   4
#define HDIM    64
#define HDTOT   256          // NHEAD*HDIM == DIN
#define NEG_SLOPE 0.2f

typedef float v2f __attribute__((ext_vector_type(2)));
typedef float v8f __attribute__((ext_vector_type(8)));

// padded LDS strides (floats): 260*4 % 64banks == 4, 516*4 % 64 == 4 -> conflict-free
#define FSTR 260
#define HSTR 516

// =====================================================================
// Fused 2-layer MLP for one path:  ft_row = relu(relu(feat@W1^T)@W2^T)
// One block = 128 threads = 4 waves, Mtile = 16 nodes.
// Layer1: wave w computes hidden cols [w*128, w*128+128)  (K=256)
// Layer2: wave w computes output cols [w*64,  w*64+64)    (K=512, from LDS)
// rearMode==0: write all rows; rearMode==1: write only rows with rear[row]!=0
// =====================================================================
__global__ __launch_bounds__(128)
void mlp_path_kernel(const float* __restrict__ feat,
                     const float* __restrict__ W1,   // [HT,DIN] row-major
                     const float* __restrict__ W2,   // [HDTOT,HT] row-major
                     const unsigned char* __restrict__ rear,
                     float* __restrict__ ft,         // [N, HDTOT]
                     int rearMode)
{
    __shared__ float sFeat[16 * FSTR];
    __shared__ float sHid [16 * HSTR];

    const int tid  = threadIdx.x;
    const int wave = tid >> 5;
    const int lane = tid & 31;
    const int m0   = blockIdx.x * 16;

    // ---- stage 16x256 feat tile into LDS (vectorized, coalesced) ----
    for (int idx = tid; idx < 16 * 64; idx += 128) {
        int row = idx >> 6;
        int c   = (idx & 63) << 2;
        float4 v = *(const float4*)(feat + (size_t)(m0 + row) * DIN + c);
        *(float4*)(&sFeat[row * FSTR + c]) = v;
    }
    __syncthreads();

    // f32 WMMA A-fragment addressing: lanes 0-15 -> rows, hold K={k0,k0+1};
    // lanes 16-31 -> same rows, K={k0+2,k0+3}
    const int r  = lane & 15;
    const int kh = (lane >> 4) << 1;   // 0 or 2
    const int nc = lane & 15;          // column within 16-wide N tile

    // ---------------- layer 1 ----------------
    for (int t = 0; t < 8; ++t) {
        const int n0   = wave * 128 + t * 16;
        const int ncol = n0 + nc;
        const float* wrow = W1 + (size_t)ncol * DIN;   // B[k][n] = W1[n][k]
        v8f acc = {};
#pragma unroll 4
        for (int k0 = 0; k0 < DIN; k0 += 4) {
            v2f a = *(const v2f*)(&sFeat[r * FSTR + k0 + kh]);
            v2f b = *(const v2f*)(wrow + k0 + kh);
            acc = __builtin_amdgcn_wmma_f32_16x16x4_f32(
                      false, a, false, b, (short)0, acc, false, false);
        }
        // C layout: VGPR j holds row (j + 8*(lane>=16)), col = lane&15
#pragma unroll
        for (int j = 0; j < 8; ++j) {
            int row = j + ((lane >> 4) << 3);
            sHid[row * HSTR + ncol] = fmaxf(acc[j], 0.0f);
        }
    }
    __syncthreads();

    // ---------------- layer 2 ----------------
    for (int t = 0; t < 4; ++t) {
        const int n0   = wave * 64 + t * 16;
        const int ncol = n0 + nc;
        const float* wrow = W2 + (size_t)ncol * HT;
        v8f acc = {};
#pragma unroll 4
        for (int k0 = 0; k0 < HT; k0 += 4) {
            v2f a = *(const v2f*)(&sHid[r * HSTR + k0 + kh]);
            v2f b = *(const v2f*)(wrow + k0 + kh);
            acc = __builtin_amdgcn_wmma_f32_16x16x4_f32(
                      false, a, false, b, (short)0, acc, false, false);
        }
#pragma unroll
        for (int j = 0; j < 8; ++j) {
            int row = m0 + j + ((lane >> 4) << 3);
            float v = fmaxf(acc[j], 0.0f);
            if (!rearMode || rear[row])
                ft[(size_t)row * HDTOT + ncol] = v;
        }
    }
}

// =====================================================================
// el[n,h] = sum_d ft[n,h,d]*attn_l[h,d] ; er likewise. One wave per node.
// =====================================================================
__global__ __launch_bounds__(256)
void attn_dot_kernel(const float* __restrict__ ft,
                     const float* __restrict__ attn_l,
                     const float* __restrict__ attn_r,
                     float* __restrict__ el, float* __restrict__ er)
{
    const int node = blockIdx.x * 8 + (threadIdx.x >> 5);
    const int lane = threadIdx.x & 31;
    const int h     = lane >> 3;            // 4 heads, 8 lanes each
    const int dbase = (lane & 7) << 3;      // 8 elements per lane
    const float* frow = ft + (size_t)node * HDTOT + h * HDIM + dbase;
    const float* al   = attn_l + h * HDIM + dbase;
    const float* ar   = attn_r + h * HDIM + dbase;
    float pl = 0.f, pr = 0.f;
#pragma unroll
    for (int j = 0; j < 8; ++j) {
        float f = frow[j];
        pl += f * al[j];
        pr += f * ar[j];
    }
    pl += __shfl_xor(pl, 1); pr += __shfl_xor(pr, 1);
    pl += __shfl_xor(pl, 2); pr += __shfl_xor(pr, 2);
    pl += __shfl_xor(pl, 4); pr += __shfl_xor(pr, 4);
    if ((lane & 7) == 0) {
        el[node * NHEAD + h] = pl;
        er[node * NHEAD + h] = pr;
    }
}

// ---- order-preserving float<->uint encoding for atomic max ----
__device__ __forceinline__ unsigned enc_f32(float x) {
    unsigned u = __float_as_uint(x);
    return (u & 0x80000000u) ? ~u : (u | 0x80000000u);
}
__device__ __forceinline__ float dec_f32(unsigned u) {
    unsigned b = (u & 0x80000000u) ? (u & 0x7FFFFFFFu) : ~u;
    return __uint_as_float(b);
}

__global__ void init_kernel(float* __restrict__ rst,
                            unsigned* __restrict__ emax_enc,
                            float* __restrict__ denom)
{
    size_t i = (size_t)blockIdx.x * blockDim.x + threadIdx.x;
    if (i < (size_t)N_NODES * HDTOT) rst[i] = 0.0f;
    if (i < (size_t)N_NODES * NHEAD) {
        emax_enc[i] = enc_f32(-__builtin_inff());
        denom[i]    = 0.0f;
    }
}

__device__ __forceinline__ float edge_e(const float* el, const float* er,
                                        int s, int d, int h)
{
    float x = el[s * NHEAD + h] + er[d * NHEAD + h];
    return x > 0.0f ? x : NEG_SLOPE * x;
}

__global__ void edge_max_kernel(const int* __restrict__ src, const int* __restrict__ dst,
                                const float* __restrict__ el, const float* __restrict__ er,
                                unsigned* __restrict__ emax_enc)
{
    int i = blockIdx.x * blockDim.x + threadIdx.x;   // exactly E*NHEAD threads
    int e = i >> 2, h = i & 3;
    int s = src[e], d = dst[e];
    float x = edge_e(el, er, s, d, h);
    atomicMax(&emax_enc[d * NHEAD + h], enc_f32(x));
}

__global__ void decode_max_kernel(const unsigned* __restrict__ emax_enc,
                                  float* __restrict__ emaxf)
{
    int i = blockIdx.x * blockDim.x + threadIdx.x;
    if (i < N_NODES * NHEAD) {
        float f = dec_f32(emax_enc[i]);
        emaxf[i] = isfinite(f) ? f : 0.0f;   // nodes with no in-edges -> 0
    }
}

__global__ void edge_denom_kernel(const int* __restrict__ src, const int* __restrict__ dst,
                                  const float* __restrict__ el, const float* __restrict__ er,
                                  const float* __restrict__ emaxf, float* __restrict__ denom)
{
    int i = blockIdx.x * blockDim.x + threadIdx.x;
    int e = i >> 2, h = i & 3;
    int s = src[e], d = dst[e];
    float x  = edge_e(el, er, s, d, h);
    float ee = expf(x - emaxf[d * NHEAD + h]);
    unsafeAtomicAdd(&denom[d * NHEAD + h], ee);
}

// one 256-thread block handles 16 edges: lane t covers (h=t/64, d=t%64)
#define EPB 16
__global__ __launch_bounds__(256)
void edge_aggregate_kernel(const int* __restrict__ src, const int* __restrict__ dst,
                           const float* __restrict__ el, const float* __restrict__ er,
                           const float* __restrict__ emaxf, const float* __restrict__ denom,
                           const float* __restrict__ ft, float* __restrict__ rst)
{
    const int t = threadIdx.x;
    const int h = t >> 6;
    const int e0 = blockIdx.x * EPB;
    for (int k = 0; k < EPB; ++k) {
        int e = e0 + k;
        int s = src[e], d = dst[e];
        float x = edge_e(el, er, s, d, h);
        float a = expf(x - emaxf[d * NHEAD + h]) /
                  fmaxf(denom[d * NHEAD + h], 1e-9f);
        float val = ft[(size_t)s * HDTOT + t] * a;     // coalesced 1KB gather
        unsafeAtomicAdd(&rst[(size_t)d * HDTOT + t], val); // L2-resident atomics
    }
}

// relu(rst) in place + overwrite the ft scratch region with the broadcast
// index outputs (index_rear, index_other) as floats.
__global__ void finalize_kernel(float* __restrict__ out,
                                const unsigned char* __restrict__ index1)
{
    const size_t NHD = (size_t)N_NODES * HDTOT;
    size_t i = (size_t)blockIdx.x * blockDim.x + threadIdx.x;  // exactly N*DIN
    out[i] = fmaxf(out[i], 0.0f);
    int n = (int)(i >> 8);
    float r = index1[n] ? 1.0f : 0.0f;
    out[NHD + i]     = r;          // index_rear
    out[2 * NHD + i] = 1.0f - r;   // index_other
}

// =====================================================================
extern "C" void kernel_launch(void* const* d_in, const int* in_sizes, int n_in,
                              void* d_out, int out_size, void* d_ws, size_t ws_size,
                              hipStream_t stream)
{
    (void)in_sizes; (void)n_in; (void)out_size; (void)ws_size;
    const float* feat      = (const float*)d_in[0];
    const float* W_h1      = (const float*)d_in[1];
    const float* W_fc      = (const float*)d_in[2];
    const float* W_h1_rear = (const float*)d_in[3];
    const float* W_fc_rear = (const float*)d_in[4];
    const float* attn_l    = (const float*)d_in[5];
    const float* attn_r    = (const float*)d_in[6];
    const unsigned char* index1 = (const unsigned char*)d_in[7]; // bool bytes
    const int* src = (const int*)d_in[9];
    const int* dst = (const int*)d_in[10];

    float* out = (float*)d_out;
    const size_t NHD = (size_t)N_NODES * HDTOT;
    float* rst = out;             // output slot 0
    float* ft  = out + NHD;       // temporarily parks ft in the index_rear slot

    // small workspace: 5 arrays of N*NHEAD
    float*    el       = (float*)d_ws;
    float*    er       = el + (size_t)N_NODES * NHEAD;
    unsigned* emax_enc = (unsigned*)(er + (size_t)N_NODES * NHEAD);
    float*    emaxf    = (float*)(emax_enc + (size_t)N_NODES * NHEAD);
    float*    denom    = emaxf + (size_t)N_NODES * NHEAD;

    init_kernel<<<(N_NODES * HDTOT) / 256, 256, 0, stream>>>(rst, emax_enc, denom);

    mlp_path_kernel<<<N_NODES / 16, 128, 0, stream>>>(feat, W_h1, W_fc, index1, ft, 0);
    mlp_path_kernel<<<N_NODES / 16, 128, 0, stream>>>(feat, W_h1_rear, W_fc_rear, index1, ft, 1);

    attn_dot_kernel<<<N_NODES / 8, 256, 0, stream>>>(ft, attn_l, attn_r, el, er);

    edge_max_kernel<<<(E_EDGES * NHEAD) / 256, 256, 0, stream>>>(src, dst, el, er, emax_enc);
    decode_max_kernel<<<(N_NODES * NHEAD + 255) / 256, 256, 0, stream>>>(emax_enc, emaxf);
    edge_denom_kernel<<<(E_EDGES * NHEAD) / 256, 256, 0, stream>>>(src, dst, el, er, emaxf, denom);
    edge_aggregate_kernel<<<E_EDGES / EPB, 256, 0, stream>>>(src, dst, el, er, emaxf, denom, ft, rst);

    finalize_kernel<<<(N_NODES * HDTOT) / 256, 256, 0, stream>>>(out, index1);
}